// SRLEmbeddings_36155034698202
// MI455X (gfx1250) — compile-verified
//
#include <hip/hip_runtime.h>
#include <hip/hip_bf16.h>

// Problem constants (from the reference)
#define BQ 32
#define SQ 32
#define LQ 64
#define DQ 768
#define AQ 10
#define TQ 4

// LDS layout strides (dwords), padded to avoid bank conflicts
#define ESTRIDE 772   // 768 + 4 pad; 772 % 64 == 4
#define WSTRIDE 68    // 64 + 4 pad;  68 % 64 == 4

typedef __attribute__((ext_vector_type(2))) float v2f;
typedef __attribute__((ext_vector_type(8))) float v8f;

__global__ void __launch_bounds__(256)
srl_embeddings_kernel(const float* __restrict__ emb,
                      const int* __restrict__ sid,
                      const int* __restrict__ amask,
                      const int* __restrict__ pred,
                      const int* __restrict__ arg0,
                      const int* __restrict__ arg1,
                      float* __restrict__ out)
{
    extern __shared__ char smem_raw[];
    float* E   = (float*)smem_raw;                 // [LQ][ESTRIDE]
    float* W   = E + LQ * ESTRIDE;                 // [32][WSTRIDE]
    int*   ids = (int*)(W + 32 * WSTRIDE);         // [LQ]
    int*   msk = ids + LQ;                         // [LQ]

    const int bs   = blockIdx.x;                   // 0..1023 == b*SQ + s
    const size_t bsz = (size_t)bs;
    const int tid  = threadIdx.x;

    // ---------------- Phase 1: async DMA of E slab (64 x 768 f32) to LDS ----
    {
        const char* gbase = (const char*)(emb + bsz * (size_t)(LQ * DQ));
        const unsigned eBase = (unsigned)(size_t)E;   // low 32 bits = LDS offset
        // 12288 x 16B chunks, 192 chunks per row of 768 floats
        for (int c = tid; c < (LQ * DQ) / 4; c += 256) {
            const int r  = c / 192;
            const int cc = c - r * 192;
            unsigned lds_addr = eBase + (unsigned)(r * (ESTRIDE * 4) + cc * 16);
            const char* gaddr = gbase + (size_t)r * (DQ * 4) + (size_t)cc * 16;
            asm volatile("global_load_async_to_lds_b128 %0, %1, off"
                         :: "v"(lds_addr), "v"(gaddr)
                         : "memory");
        }
    }

    // ---------------- Phase 2: stage sentence ids / mask -------------------
    if (tid < LQ) {
        ids[tid] = sid[bsz * LQ + tid];
        msk[tid] = amask[bsz * LQ + tid];
    }
    __syncthreads();

    // ---------------- Phase 3: build the 32 x 64 weight matrix -------------
    if (tid < 32) {
        float* wrow = W + tid * WSTRIDE;
        if (tid == 0) {
            int cnt = 0;
            for (int l = 0; l < LQ; ++l) cnt += msk[l];
            const float inv = 1.0f / (float)(cnt > 0 ? cnt : 1);
            for (int l = 0; l < LQ; ++l) wrow[l] = msk[l] ? inv : 0.0f;
        } else if (tid <= 30) {
            const int idx = tid - 1;
            const int ty  = idx / AQ;        // 0=pred, 1=arg0, 2=arg1
            const int a   = idx % AQ;
            const int* ap = (ty == 0 ? pred : (ty == 1 ? arg0 : arg1))
                          + (bsz * AQ + a) * TQ;
            int selId = 0, selCnt = 0;
            bool any = false;
            for (int t = 0; t < TQ; ++t) {   // last valid token wins
                const int id = ap[t];
                if (id == 0) continue;
                int c = 0;
                for (int l = 0; l < LQ; ++l) c += (ids[l] == id) ? 1 : 0;
                if (c > 0) { selId = id; selCnt = c; any = true; }
            }
            if (any) {
                const float inv = 1.0f / (float)selCnt;
                for (int l = 0; l < LQ; ++l)
                    wrow[l] = (ids[l] == selId) ? inv : 0.0f;
            } else {
                for (int l = 0; l < LQ; ++l) wrow[l] = 0.0f;
            }
        } else {  // row 31: padding row, result discarded but keep defined
            for (int l = 0; l < LQ; ++l) wrow[l] = 0.0f;
        }
    }

    // Wait for this wave's async copies; barrier covers all waves' copies + W
    asm volatile("s_wait_asynccnt 0x0" ::: "memory");
    __syncthreads();

    // ---------------- Phase 4: R[32,768] = W[32,64] x E[64,768] via WMMA ---
    const int wave  = tid >> 5;
    const int lane  = tid & 31;
    const int lhalf = lane >> 4;   // 0: lanes 0-15, 1: lanes 16-31
    const int l15   = lane & 15;

    float* const outAvg = out + bsz * (size_t)DQ;
    float* const outArgBase = out + (size_t)BQ * SQ * DQ;
    const size_t argSect = (size_t)BQ * SQ * AQ * DQ;

    // 96 tiles of 16x16 (Mtiles=2, Ntiles=48), 12 per wave
    for (int tile = wave; tile < 96; tile += 8) {
        const int mt = tile & 1;
        const int n0 = (tile >> 1) * 16;
        v8f acc = {};
#pragma unroll
        for (int k = 0; k < LQ; k += 4) {
            // A fragment (16x4 f32): VGPR0 = {K, K+2}, VGPR1 = {K+1, K+3}
            const float* wr = W + (mt * 16 + l15) * WSTRIDE + (k + 2 * lhalf);
            v2f af; af.x = wr[0]; af.y = wr[1];
            // B fragment (4x16 f32): lane&15 = N, lane half selects K pair
            const float* er = E + (k + 2 * lhalf) * ESTRIDE + n0 + l15;
            v2f bf; bf.x = er[0]; bf.y = er[ESTRIDE];
            acc = __builtin_amdgcn_wmma_f32_16x16x4_f32(
                /*neg_a=*/false, af, /*neg_b=*/false, bf,
                /*c_mod=*/(short)0, acc, /*reuse_a=*/false, /*reuse_b=*/false);
        }
        // C/D layout: VGPR r -> M = r (lanes 0-15) / r+8 (lanes 16-31)
        // Outputs are written once and never re-read: use non-temporal stores
        // so the 97.5 MB result stream does not evict the streaming inputs.
#pragma unroll
        for (int r = 0; r < 8; ++r) {
            const int row = mt * 16 + lhalf * 8 + r;
            if (row == 0) {
                __builtin_nontemporal_store(acc[r], outAvg + n0 + l15);
            } else if (row < 31) {
                const int ty = (row - 1) / AQ;
                const int a  = (row - 1) % AQ;
                float* dst = outArgBase + (size_t)ty * argSect
                           + (bsz * AQ + a) * (size_t)DQ;
                __builtin_nontemporal_store(acc[r], dst + n0 + l15);
            }
        }
    }
}

extern "C" void kernel_launch(void* const* d_in, const int* in_sizes, int n_in,
                              void* d_out, int out_size, void* d_ws, size_t ws_size,
                              hipStream_t stream) {
    (void)in_sizes; (void)n_in; (void)out_size; (void)d_ws; (void)ws_size;
    const float* emb   = (const float*)d_in[0];
    const int*   sids  = (const int*)d_in[1];
    const int*   masks = (const int*)d_in[2];
    const int*   pred  = (const int*)d_in[3];
    const int*   a0    = (const int*)d_in[4];
    const int*   a1    = (const int*)d_in[5];
    float*       out   = (float*)d_out;

    const size_t smem = (size_t)(LQ * ESTRIDE + 32 * WSTRIDE) * sizeof(float)
                      + (size_t)(2 * LQ) * sizeof(int);   // ~207 KB (<320 KB/WGP)
    (void)hipFuncSetAttribute((const void*)srl_embeddings_kernel,
                              hipFuncAttributeMaxDynamicSharedMemorySize, (int)smem);

    srl_embeddings_kernel<<<dim3(BQ * SQ), dim3(256), smem, stream>>>(
        emb, sids, masks, pred, a0, a1, out);
}